// CustomRNN_3075196584540
// MI455X (gfx1250) — compile-verified
//
#include <hip/hip_runtime.h>
#include <math.h>

// Problem sizes (fixed by the reference)
constexpr int BB = 64;      // batch
constexpr int TT = 512;     // time steps
constexpr int DD = 512;     // input dim
constexpr int HH = 1024;    // hidden dim
constexpr int GG = 4 * HH;  // 4096 gate width

typedef float v2f __attribute__((ext_vector_type(2)));
typedef float v8f __attribute__((ext_vector_type(8)));

__device__ __forceinline__ v8f wmma_f32_k4(v2f a, v2f b, v8f c) {
  // V_WMMA_F32_16X16X4_F32 : D = A(16x4) * B(4x16) + C, fp32 exact
  return __builtin_amdgcn_wmma_f32_16x16x4_f32(
      /*neg_a=*/false, a, /*neg_b=*/false, b,
      /*c_mod=*/(short)0, c, /*reuse_a=*/false, /*reuse_b=*/false);
}

__device__ __forceinline__ float sigmoidf_(float x) {
  return 1.0f / (1.0f + __expf(-x));
}

// ---------------------------------------------------------------------------
// Kernel 0: zero h0 + h1 + c state in workspace
// ---------------------------------------------------------------------------
__global__ void zero_ws_kernel(float* __restrict__ p, int n) {
  int i = blockIdx.x * blockDim.x + threadIdx.x;
  if (i < n) p[i] = 0.0f;
}

// ---------------------------------------------------------------------------
// Kernel 1: xg[b*T+t, 4H] = x[b*T+t, :D] @ Wx[D,4H] + bias   (fully parallel)
// grid = (GG/64, M/(16*8)) = (64, 256), block = 256 (8 waves)
// Each wave: one 16-row M tile x 64 columns (4 independent 16x16 WMMA tiles,
// A fragment reused across the 4 N tiles to break the accumulator RAW chain).
// ---------------------------------------------------------------------------
__global__ __launch_bounds__(256)
void xg_gemm_kernel(const float* __restrict__ x, const float* __restrict__ Wx,
                    const float* __restrict__ bias, float* __restrict__ xg) {
  const int lane = threadIdx.x & 31;
  const int wv   = threadIdx.x >> 5;          // 0..7
  const int ln   = lane & 15;
  const int hi   = lane >> 4;                 // 0 or 1
  const int koff = hi << 1;                   // 0 or 2 (A/B K sub-offset)

  const int m0 = (blockIdx.y * 8 + wv) * 16;  // row tile base in [0, 32768)
  const int nb = blockIdx.x * 64;             // column group base in [0, 4096)

  const float* xrow = x + (size_t)(m0 + ln) * DD;

  v8f acc[4];
  #pragma unroll
  for (int j = 0; j < 4; ++j) acc[j] = (v8f){0,0,0,0,0,0,0,0};

  for (int k0 = 0; k0 < DD; k0 += 4) {
    // A fragment: lane ln holds row m0+ln, k = k0+koff, k0+koff+1 (aligned f64 load)
    v2f a = *(const v2f*)(xrow + k0 + koff);
    #pragma unroll
    for (int j = 0; j < 4; ++j) {
      const int col = nb + j * 16 + ln;
      v2f b;
      b.x = Wx[(size_t)(k0 + koff    ) * GG + col];
      b.y = Wx[(size_t)(k0 + koff + 1) * GG + col];
      acc[j] = wmma_f32_k4(a, b, acc[j]);
    }
  }

  // C/D layout: VGPR v -> row (v + 8*hi), col = ln
  #pragma unroll
  for (int j = 0; j < 4; ++j) {
    const int col = nb + j * 16 + ln;
    const float bc = bias[col];
    #pragma unroll
    for (int v = 0; v < 8; ++v) {
      const int row = m0 + v + (hi << 3);
      xg[(size_t)row * GG + col] = acc[j][v] + bc;
    }
  }
}

// ---------------------------------------------------------------------------
// Kernel 2: one LSTM time step.
//   gates[B,4H] = xg_t (or x_t@Wx + b) + h_prev @ Wh ;  c,h update fused.
// grid = (HH/16, BB/16) = (64, 4), block = 128 (4 waves, one per gate).
// Each wave computes a 16x16 gate tile with fp32 WMMA over K=1024 (Wh is
// L2-resident: 16MB << 192MB L2), stages it in LDS, then the block applies
// the cell nonlinearity for its unique (16 rows x 16 cols) slice.
// h is ping-ponged across steps to avoid cross-block RAW hazards.
// ---------------------------------------------------------------------------
template <bool USE_XG>
__global__ __launch_bounds__(128)
void lstm_step_kernel(const float* __restrict__ xg,   // [B*T, 4H] or null
                      const float* __restrict__ x,    // [B, T, D]  (fused path)
                      const float* __restrict__ Wx,   // [D, 4H]    (fused path)
                      const float* __restrict__ Wh,   // [H, 4H]
                      const float* __restrict__ bias, // [4H]       (fused path)
                      const float* __restrict__ hin,  // [B, H]
                      float* __restrict__ hout,       // [B, H]
                      float* __restrict__ cbuf,       // [B, H]
                      float* __restrict__ out,        // [B, T, H]
                      int t) {
  __shared__ float gsm[4][16][16];  // [gate][row][col]

  const int lane = threadIdx.x & 31;
  const int gate = threadIdx.x >> 5;          // 0..3 : i,f,g,o
  const int ln   = lane & 15;
  const int hi   = lane >> 4;
  const int koff = hi << 1;

  const int m0 = blockIdx.y * 16;             // batch-row tile
  const int n0 = blockIdx.x * 16;             // hidden-column tile
  const int col = gate * HH + n0 + ln;        // column in 4H gate space

  const float* hrow = hin + (size_t)(m0 + ln) * HH;

  // Two independent accumulators to break the WMMA->WMMA RAW chain.
  v8f accA = (v8f){0,0,0,0,0,0,0,0};
  v8f accB = (v8f){0,0,0,0,0,0,0,0};

  for (int k0 = 0; k0 < HH; k0 += 8) {
    v2f a0 = *(const v2f*)(hrow + k0 + koff);
    v2f b0;
    b0.x = Wh[(size_t)(k0 + koff    ) * GG + col];
    b0.y = Wh[(size_t)(k0 + koff + 1) * GG + col];
    accA = wmma_f32_k4(a0, b0, accA);

    v2f a1 = *(const v2f*)(hrow + k0 + 4 + koff);
    v2f b1;
    b1.x = Wh[(size_t)(k0 + 4 + koff    ) * GG + col];
    b1.y = Wh[(size_t)(k0 + 4 + koff + 1) * GG + col];
    accB = wmma_f32_k4(a1, b1, accB);
  }

  if (!USE_XG) {
    // Fused input projection: gates += x_t @ Wx
    const float* xrow = x + ((size_t)(m0 + ln) * TT + t) * DD;
    for (int k0 = 0; k0 < DD; k0 += 8) {
      v2f a0 = *(const v2f*)(xrow + k0 + koff);
      v2f b0;
      b0.x = Wx[(size_t)(k0 + koff    ) * GG + col];
      b0.y = Wx[(size_t)(k0 + koff + 1) * GG + col];
      accA = wmma_f32_k4(a0, b0, accA);

      v2f a1 = *(const v2f*)(xrow + k0 + 4 + koff);
      v2f b1;
      b1.x = Wx[(size_t)(k0 + 4 + koff    ) * GG + col];
      b1.y = Wx[(size_t)(k0 + 4 + koff + 1) * GG + col];
      accB = wmma_f32_k4(a1, b1, accB);
    }
  }

  v8f acc = accA + accB;

  const float bc = USE_XG ? 0.0f : bias[col];
  #pragma unroll
  for (int v = 0; v < 8; ++v) {
    const int r = v + (hi << 3);              // local row 0..15
    float g = acc[v] + bc;
    if (USE_XG) {
      g += xg[((size_t)(m0 + r) * TT + t) * GG + col];
    }
    gsm[gate][r][ln] = g;
  }

  __syncthreads();

  // Elementwise cell update: 256 elements, 128 threads.
  for (int e = threadIdx.x; e < 256; e += 128) {
    const int r = e >> 4;
    const int c = e & 15;
    const float gi = gsm[0][r][c];
    const float gf = gsm[1][r][c];
    const float gg = gsm[2][r][c];
    const float go = gsm[3][r][c];

    const float i = sigmoidf_(gi);
    const float f = sigmoidf_(gf);
    const float g = tanhf(gg);
    const float o = sigmoidf_(go);

    const int b    = m0 + r;
    const int hidx = n0 + c;
    const size_t ci = (size_t)b * HH + hidx;

    const float cn = f * cbuf[ci] + i * g;
    cbuf[ci] = cn;
    const float hn = o * tanhf(cn);
    hout[ci] = hn;
    out[((size_t)b * TT + t) * HH + hidx] = hn;
  }
}

// ---------------------------------------------------------------------------
// Host launcher
// ---------------------------------------------------------------------------
extern "C" void kernel_launch(void* const* d_in, const int* in_sizes, int n_in,
                              void* d_out, int out_size, void* d_ws, size_t ws_size,
                              hipStream_t stream) {
  (void)in_sizes; (void)n_in; (void)out_size;

  const float* x    = (const float*)d_in[0];  // [B, T, D]
  const float* Wx   = (const float*)d_in[1];  // [D, 4H]
  const float* Wh   = (const float*)d_in[2];  // [H, 4H]
  const float* bias = (const float*)d_in[3];  // [4H]
  float* out = (float*)d_out;                 // [B, T, H]

  // Workspace layout (floats): h0 | h1 | c | xg(optional)
  constexpr size_t BH = (size_t)BB * HH;      // 65536
  float* ws = (float*)d_ws;
  float* h0 = ws;
  float* h1 = ws + BH;
  float* cb = ws + 2 * BH;
  float* xg = ws + 3 * BH;

  const size_t state_floats = 3 * BH;
  const size_t xg_floats    = (size_t)BB * TT * GG;  // 128M floats = 512MB
  const bool use_xg = ws_size >= (state_floats + xg_floats) * sizeof(float);

  // Re-initialize recurrent state every call (graph replays must be identical).
  {
    const int n = (int)state_floats;
    zero_ws_kernel<<<(n + 255) / 256, 256, 0, stream>>>(ws, n);
  }

  if (use_xg) {
    // Fully parallel input projection: [32768,512] x [512,4096] + bias
    dim3 grid(GG / 64, (BB * TT) / (16 * 8));  // (64, 256)
    xg_gemm_kernel<<<grid, 256, 0, stream>>>(x, Wx, bias, xg);
  }

  dim3 sgrid(HH / 16, BB / 16);  // (64, 4)
  for (int t = 0; t < TT; ++t) {
    const float* hin = (t & 1) ? h1 : h0;
    float*       hou = (t & 1) ? h0 : h1;
    if (use_xg) {
      lstm_step_kernel<true><<<sgrid, 128, 0, stream>>>(
          xg, nullptr, nullptr, Wh, nullptr, hin, hou, cb, out, t);
    } else {
      lstm_step_kernel<false><<<sgrid, 128, 0, stream>>>(
          nullptr, x, Wx, Wh, bias, hin, hou, cb, out, t);
    }
  }
}